// Seq2SeqGRUWithAttention_11115375362493
// MI455X (gfx1250) — compile-verified
//
#include <hip/hip_runtime.h>
#include <math.h>

typedef __attribute__((ext_vector_type(2))) float v2f;
typedef __attribute__((ext_vector_type(8))) float v8f;
typedef __attribute__((ext_vector_type(4))) int   v4i;

#define B_  32
#define S_  128
#define T_  128
#define E_  512
#define H_  512
#define VT_ 32000

#define KC_   32          // K chunk staged in LDS (big GEMM)
#define LDAS_ 36          // LDS row stride (floats): %4==0 for b128, not %32 -> no bank conflicts

// gfx1250 async global->LDS staging path (ASYNCcnt), with safe fallback.
#if defined(__AMDGCN__) && __has_builtin(__builtin_amdgcn_global_load_async_to_lds_b128) && \
    __has_builtin(__builtin_amdgcn_s_wait_asynccnt)
#define ASYNC_LDS 1
#define AS_GLOBAL __attribute__((address_space(1)))
#define AS_LOCAL  __attribute__((address_space(3)))
#else
#define ASYNC_LDS 0
#endif

__device__ __forceinline__ float sigmoidf_(float x) {
    return 1.0f / (1.0f + __expf(-x));
}

// ---------------------------------------------------------------------------
// Small fp32 WMMA GEMM (latency-bound recurrent steps, M == 32):
// one wave = one 16x16 tile, 4 waves/block. Maximizes wave parallelism.
// Requires M%16==0, N%64==0, K%4==0.
// ---------------------------------------------------------------------------
__global__ __launch_bounds__(128)
void wmma_gemm_f32_small(const float* __restrict__ A, const float* __restrict__ Bm,
                         const float* __restrict__ bias, float* __restrict__ C,
                         int M, int N, int K, int lda, int ldb, int ldc)
{
    const int lane = threadIdx.x & 31;
    const int wave = threadIdx.x >> 5;
    const int m0   = blockIdx.y << 4;
    const int n0   = (blockIdx.x << 6) + (wave << 4);
    if (m0 >= M || n0 >= N) return;          // uniform per wave: EXEC stays full

    const int half = lane >> 4;
    const int l15  = lane & 15;

    const float* arow = A  + (size_t)(m0 + l15) * lda + (half << 1);
    const float* bcol = Bm + (size_t)(half << 1) * ldb + (n0 + l15);

    v8f acc = {};
#pragma unroll 2
    for (int k = 0; k < K; k += 4) {
        v2f a = *(const v2f*)(arow + k);
        v2f b;
        b.x = bcol[(size_t)k * ldb];
        b.y = bcol[(size_t)(k + 1) * ldb];
        acc = __builtin_amdgcn_wmma_f32_16x16x4_f32(false, a, false, b, (short)0, acc,
                                                    false, false);
    }

    const int   n  = n0 + l15;
    const float bv = bias ? bias[n] : 0.0f;
#pragma unroll
    for (int i = 0; i < 8; ++i) {
        const int m = m0 + i + (half << 3);
        C[(size_t)m * ldc + n] = acc[i] + bv;
    }
}

// ---------------------------------------------------------------------------
// Big fp32 WMMA GEMM (throughput-bound, M >= 32):
// block = 4 waves, block tile 32(M) x 256(N); each wave 32x64 (2 M-tiles x
// 4 N-tiles, 8 accumulators). A chunk [32 x KC_] double-buffered in LDS:
// staged via GLOBAL_LOAD_ASYNC_TO_LDS_B128 (ASYNCcnt) when available, so the
// next chunk streams in while the current chunk's 64 WMMAs execute.
// Requires M%32==0, N%256==0, K%KC_==0.
// ---------------------------------------------------------------------------
__global__ __launch_bounds__(128)
void wmma_gemm_f32_big(const float* __restrict__ A, const float* __restrict__ Bm,
                       const float* __restrict__ bias, float* __restrict__ C,
                       int M, int N, int K, int lda, int ldb, int ldc)
{
    __shared__ float sA[2][32 * LDAS_];

    const int tid  = threadIdx.x;
    const int lane = tid & 31;
    const int wave = tid >> 5;
    const int m0   = blockIdx.y << 5;                  // 32 rows / block
    const int nw   = (blockIdx.x << 8) + (wave << 6);  // 64 cols / wave
    const int half = lane >> 4;
    const int l15  = lane & 15;

    // A-staging coordinates: 128 threads cover 32 rows x 32 cols (8 floats each)
    const int sr = tid >> 2;           // 0..31
    const int sc = (tid & 3) << 3;     // 0,8,16,24

    const float* abase = A  + (size_t)(m0 + sr) * lda + sc;
    const float* bbase = Bm + (size_t)(half << 1) * ldb + nw + l15;
    float*       lbase = &sA[0][sr * LDAS_ + sc];      // buffer 0 staging slot

    v8f acc[2][4] = {};

#if ASYNC_LDS
    // async: memory -> LDS without a VGPR round-trip, tracked by ASYNCcnt
#define STAGE_A(sel, kcv)                                                            \
    do {                                                                             \
        const float* gp_ = abase + (kcv);                                            \
        float*       lp_ = lbase + (sel) * (32 * LDAS_);                             \
        __builtin_amdgcn_global_load_async_to_lds_b128(                              \
            (AS_GLOBAL v4i*)(gp_), (AS_LOCAL v4i*)(lp_), 0, 0);                      \
        __builtin_amdgcn_global_load_async_to_lds_b128(                              \
            (AS_GLOBAL v4i*)((gp_) + 4), (AS_LOCAL v4i*)((lp_) + 4), 0, 0);          \
    } while (0)
#define STAGE_WAIT() __builtin_amdgcn_s_wait_asynccnt(0)
#else
#define STAGE_A(sel, kcv)                                                            \
    do {                                                                             \
        const float* gp_ = abase + (kcv);                                            \
        float*       lp_ = lbase + (sel) * (32 * LDAS_);                             \
        float4 p0_ = *(const float4*)(gp_);                                          \
        float4 p1_ = *(const float4*)((gp_) + 4);                                    \
        *(float4*)(lp_)       = p0_;                                                 \
        *(float4*)((lp_) + 4) = p1_;                                                 \
    } while (0)
#define STAGE_WAIT() ((void)0)
#endif

    // prologue: stage chunk 0 into buffer 0
    STAGE_A(0, 0);

    int buf = 0;
    for (int kc = 0; kc < K; kc += KC_) {
        STAGE_WAIT();                          // drain this wave's async stages
        __syncthreads();                       // all waves' staged data visible

        if (kc + KC_ < K)
            STAGE_A(buf ^ 1, kc + KC_);        // overlap next chunk with compute

        const float* sAb = &sA[buf][0];
#pragma unroll
        for (int k = 0; k < KC_; k += 4) {
            const int kb = k + (half << 1);
            v2f a0 = *(const v2f*)&sAb[l15 * LDAS_ + kb];
            v2f a1 = *(const v2f*)&sAb[(l15 + 16) * LDAS_ + kb];
            const float* brow0 = bbase + (size_t)(kc + k) * ldb;
            const float* brow1 = brow0 + ldb;
#pragma unroll
            for (int j = 0; j < 4; ++j) {
                v2f b;
                b.x = brow0[j << 4];
                b.y = brow1[j << 4];
                acc[0][j] = __builtin_amdgcn_wmma_f32_16x16x4_f32(false, a0, false, b,
                                                                 (short)0, acc[0][j],
                                                                 false, false);
                acc[1][j] = __builtin_amdgcn_wmma_f32_16x16x4_f32(false, a1, false, b,
                                                                 (short)0, acc[1][j],
                                                                 false, false);
            }
        }
        buf ^= 1;
    }

    const int nbase = nw + l15;
#pragma unroll
    for (int mt = 0; mt < 2; ++mt) {
#pragma unroll
        for (int j = 0; j < 4; ++j) {
            const int   n  = nbase + (j << 4);
            const float bv = bias ? bias[n] : 0.0f;
#pragma unroll
            for (int i = 0; i < 8; ++i) {
                const int m = m0 + (mt << 4) + i + (half << 3);
                C[(size_t)m * ldc + n] = acc[mt][j][i] + bv;
            }
        }
    }
#undef STAGE_A
#undef STAGE_WAIT
}

// ---------------------------------------------------------------------------
// Elementwise / helper kernels
// ---------------------------------------------------------------------------
__global__ void zero_f32(float* __restrict__ p, int n)
{
    int i = blockIdx.x * blockDim.x + threadIdx.x;
    if (i < n) p[i] = 0.0f;
}

__global__ void gather_rows(const float* __restrict__ table, const int* __restrict__ idx,
                            float* __restrict__ out, int rows, int cols)
{
    size_t i = (size_t)blockIdx.x * blockDim.x + threadIdx.x;
    if (i >= (size_t)rows * cols) return;
    int r = (int)(i / cols), c = (int)(i % cols);
    out[i] = table[(size_t)idx[r] * cols + c];
}

__global__ void gru_gates(const float* __restrict__ gx, int gxStride,
                          const float* __restrict__ hrz,
                          const float* __restrict__ h,
                          float* __restrict__ rh, float* __restrict__ z)
{
    int i = blockIdx.x * blockDim.x + threadIdx.x;
    if (i >= B_ * H_) return;
    int b = i / H_, j = i % H_;
    const float* g = gx + (size_t)b * gxStride;
    float r  = sigmoidf_(g[j]      + hrz[b * 2 * H_ + j]);
    float zz = sigmoidf_(g[H_ + j] + hrz[b * 2 * H_ + H_ + j]);
    rh[i] = r * h[i];
    z[i]  = zz;
}

__global__ void gru_finish(const float* __restrict__ gx, int gxStride,
                           const float* __restrict__ rhUn,
                           const float* __restrict__ z,
                           float* __restrict__ h,
                           float* __restrict__ hs_out, int hsStride)
{
    int i = blockIdx.x * blockDim.x + threadIdx.x;
    if (i >= B_ * H_) return;
    int b = i / H_, j = i % H_;
    float xn = gx[(size_t)b * gxStride + 2 * H_ + j];
    float n  = tanhf(xn + rhUn[i]);
    float hn = (1.0f - z[i]) * h[i] + z[i] * n;
    h[i] = hn;
    hs_out[(size_t)b * hsStride + j] = hn;
}

__global__ __launch_bounds__(128)
void attn_energy(const float* __restrict__ wh, const float* __restrict__ sproj,
                 const float* __restrict__ v, float* __restrict__ energy)
{
    int gw   = blockIdx.x * (blockDim.x >> 5) + (threadIdx.x >> 5);
    int lane = threadIdx.x & 31;
    if (gw >= B_ * S_) return;
    int b = gw / S_;
    const float* wr = wh + (size_t)gw * H_;
    const float* sp = sproj + (size_t)b * H_;
    float acc = 0.0f;
    for (int j = lane; j < H_; j += 32)
        acc += tanhf(wr[j] + sp[j]) * v[j];
    for (int off = 16; off >= 1; off >>= 1)
        acc += __shfl_xor(acc, off, 32);
    if (lane == 0) energy[gw] = acc;
}

__global__ void softmax_rows(const float* __restrict__ e, float* __restrict__ a)
{
    __shared__ float sm[S_];
    int b = blockIdx.x, s = threadIdx.x;
    float x = e[b * S_ + s];
    sm[s] = x; __syncthreads();
    for (int off = S_ / 2; off > 0; off >>= 1) {
        if (s < off) sm[s] = fmaxf(sm[s], sm[s + off]);
        __syncthreads();
    }
    float mx = sm[0]; __syncthreads();
    float ex = __expf(x - mx);
    sm[s] = ex; __syncthreads();
    for (int off = S_ / 2; off > 0; off >>= 1) {
        if (s < off) sm[s] += sm[s + off];
        __syncthreads();
    }
    a[b * S_ + s] = ex / sm[0];
}

__global__ void attn_ctx(const float* __restrict__ a, const float* __restrict__ hs,
                         float* __restrict__ ctx, float* __restrict__ hc_row, int hcStride)
{
    int i = blockIdx.x * blockDim.x + threadIdx.x;
    if (i >= B_ * H_) return;
    int b = i / H_, j = i % H_;
    const float* ab = a + b * S_;
    const float* hb = hs + (size_t)b * S_ * H_ + j;
    float acc = 0.0f;
    for (int s = 0; s < S_; ++s) acc += ab[s] * hb[(size_t)s * H_];
    ctx[i] = acc;
    hc_row[(size_t)b * hcStride + j] = acc;
}

__global__ void concat_x(const float* __restrict__ emb, int embStride,
                         const float* __restrict__ ctx, float* __restrict__ x)
{
    int i = blockIdx.x * blockDim.x + threadIdx.x;
    if (i >= B_ * (E_ + H_)) return;
    int b = i / (E_ + H_), j = i % (E_ + H_);
    x[i] = (j < E_) ? emb[(size_t)b * embStride + j] : ctx[b * H_ + (j - E_)];
}

__global__ __launch_bounds__(256)
void log_softmax_rows(float* __restrict__ p)
{
    __shared__ float sm[256];
    int row = blockIdx.x, tid = threadIdx.x;
    float* r = p + (size_t)row * VT_;
    float mx = -3.402823466e38f;
    for (int v = tid; v < VT_; v += 256) mx = fmaxf(mx, r[v]);
    sm[tid] = mx; __syncthreads();
    for (int off = 128; off > 0; off >>= 1) {
        if (tid < off) sm[tid] = fmaxf(sm[tid], sm[tid + off]);
        __syncthreads();
    }
    mx = sm[0]; __syncthreads();
    float sum = 0.0f;
    for (int v = tid; v < VT_; v += 256) sum += __expf(r[v] - mx);
    sm[tid] = sum; __syncthreads();
    for (int off = 128; off > 0; off >>= 1) {
        if (tid < off) sm[tid] += sm[tid + off];
        __syncthreads();
    }
    float lse = mx + __logf(sm[0]);
    for (int v = tid; v < VT_; v += 256) r[v] -= lse;
}

// ---------------------------------------------------------------------------
// Host orchestration
// ---------------------------------------------------------------------------
extern "C" void kernel_launch(void* const* d_in, const int* in_sizes, int n_in,
                              void* d_out, int out_size, void* d_ws, size_t ws_size,
                              hipStream_t stream)
{
    (void)in_sizes; (void)n_in; (void)out_size; (void)ws_size;

    const int*   src      = (const int*)  d_in[0];
    const int*   tgt      = (const int*)  d_in[1];
    const float* enc_emb  = (const float*)d_in[2];
    const float* enc_Wx   = (const float*)d_in[3];
    const float* enc_bx   = (const float*)d_in[4];
    const float* enc_Urz  = (const float*)d_in[5];
    const float* enc_brz  = (const float*)d_in[6];
    const float* enc_Un   = (const float*)d_in[7];
    const float* enc_bn   = (const float*)d_in[8];
    const float* dec_emb  = (const float*)d_in[9];
    const float* dec_Wx   = (const float*)d_in[10];
    const float* dec_bx   = (const float*)d_in[11];
    const float* dec_Urz  = (const float*)d_in[12];
    const float* dec_brz  = (const float*)d_in[13];
    const float* dec_Un   = (const float*)d_in[14];
    const float* dec_bn   = (const float*)d_in[15];
    const float* attn_Wh  = (const float*)d_in[16];
    const float* attn_Ws  = (const float*)d_in[17];
    const float* attn_v   = (const float*)d_in[18];
    const float* out_W    = (const float*)d_in[19];
    const float* out_b    = (const float*)d_in[20];
    float* out = (float*)d_out;

    // ---- workspace carve-up (all fp32) ----
    float* ws = (float*)d_ws;
    float* src_emb = ws; ws += (size_t)B_ * S_ * E_;
    float* enc_gx  = ws; ws += (size_t)B_ * S_ * 3 * H_;
    float* enc_hs  = ws; ws += (size_t)B_ * S_ * H_;
    float* wh_enc  = ws; ws += (size_t)B_ * S_ * H_;
    float* tgt_emb = ws; ws += (size_t)B_ * T_ * E_;
    float* hc_all  = ws; ws += (size_t)B_ * T_ * 2 * H_;
    float* h       = ws; ws += B_ * H_;
    float* hrz     = ws; ws += B_ * 2 * H_;
    float* rh      = ws; ws += B_ * H_;
    float* zb      = ws; ws += B_ * H_;
    float* rhUn    = ws; ws += B_ * H_;
    float* sproj   = ws; ws += B_ * H_;
    float* energy  = ws; ws += B_ * S_;
    float* attnw   = ws; ws += B_ * S_;
    float* ctx     = ws; ws += B_ * H_;
    float* xcat    = ws; ws += B_ * (E_ + H_);
    float* dec_gx  = ws; ws += B_ * 3 * H_;

    // small: latency-bound recurrent GEMMs (M == 32)
    auto gemm_s = [&](const float* A, const float* Bm, const float* bias, float* C,
                      int M, int N, int K, int lda, int ldb, int ldc) {
        dim3 grid(N / 64, M / 16);
        wmma_gemm_f32_small<<<grid, dim3(128), 0, stream>>>(A, Bm, bias, C, M, N, K, lda, ldb, ldc);
    };
    // big: throughput-bound batched GEMMs (M = B*S = 4096)
    auto gemm_b = [&](const float* A, const float* Bm, const float* bias, float* C,
                      int M, int N, int K, int lda, int ldb, int ldc) {
        dim3 grid(N / 256, M / 32);
        wmma_gemm_f32_big<<<grid, dim3(128), 0, stream>>>(A, Bm, bias, C, M, N, K, lda, ldb, ldc);
    };

    const int BH_BLK = (B_ * H_ + 255) / 256;

    // h0 = 0
    zero_f32<<<BH_BLK, 256, 0, stream>>>(h, B_ * H_);

    // ---------------- encoder ----------------
    {
        size_t tot = (size_t)B_ * S_ * E_;
        gather_rows<<<(int)((tot + 255) / 256), 256, 0, stream>>>(enc_emb, src, src_emb, B_ * S_, E_);
    }
    // enc_gx = src_emb @ enc_Wx + enc_bx   [4096, 1536]  (hoisted out of scan)
    gemm_b(src_emb, enc_Wx, enc_bx, enc_gx, B_ * S_, 3 * H_, E_, E_, 3 * H_, 3 * H_);

    for (int t = 0; t < S_; ++t) {
        gemm_s(h, enc_Urz, enc_brz, hrz, B_, 2 * H_, H_, H_, 2 * H_, 2 * H_);
        gru_gates<<<BH_BLK, 256, 0, stream>>>(enc_gx + (size_t)t * 3 * H_, S_ * 3 * H_, hrz, h, rh, zb);
        gemm_s(rh, enc_Un, enc_bn, rhUn, B_, H_, H_, H_, H_, H_);
        gru_finish<<<BH_BLK, 256, 0, stream>>>(enc_gx + (size_t)t * 3 * H_, S_ * 3 * H_, rhUn, zb,
                                               h, enc_hs + (size_t)t * H_, S_ * H_);
    }

    // wh_enc = enc_hs @ attn_Wh   [4096, 512]
    gemm_b(enc_hs, attn_Wh, nullptr, wh_enc, B_ * S_, H_, H_, H_, H_, H_);

    // ---------------- decoder ----------------
    {
        size_t tot = (size_t)B_ * T_ * E_;
        gather_rows<<<(int)((tot + 255) / 256), 256, 0, stream>>>(dec_emb, tgt, tgt_emb, B_ * T_, E_);
    }

    for (int t = 0; t < T_; ++t) {
        gemm_s(h, attn_Ws, nullptr, sproj, B_, H_, H_, H_, H_, H_);
        attn_energy<<<(B_ * S_) / 4, 128, 0, stream>>>(wh_enc, sproj, attn_v, energy);
        softmax_rows<<<B_, S_, 0, stream>>>(energy, attnw);
        attn_ctx<<<BH_BLK, 256, 0, stream>>>(attnw, enc_hs, ctx,
                                             hc_all + (size_t)t * 2 * H_ + H_, T_ * 2 * H_);
        concat_x<<<(B_ * (E_ + H_) + 255) / 256, 256, 0, stream>>>(
            tgt_emb + (size_t)t * E_, T_ * E_, ctx, xcat);
        gemm_s(xcat, dec_Wx, dec_bx, dec_gx, B_, 3 * H_, E_ + H_, E_ + H_, 3 * H_, 3 * H_);
        gemm_s(h, dec_Urz, dec_brz, hrz, B_, 2 * H_, H_, H_, 2 * H_, 2 * H_);
        gru_gates<<<BH_BLK, 256, 0, stream>>>(dec_gx, 3 * H_, hrz, h, rh, zb);
        gemm_s(rh, dec_Un, dec_bn, rhUn, B_, H_, H_, H_, H_, H_);
        gru_finish<<<BH_BLK, 256, 0, stream>>>(dec_gx, 3 * H_, rhUn, zb,
                                               h, hc_all + (size_t)t * 2 * H_, T_ * 2 * H_);
    }

    // ---------------- batched output projection (dominant 268-GFLOP GEMM) ----------------
    gemm_b(hc_all, out_W, out_b, out, B_ * T_, VT_, 2 * H_, 2 * H_, VT_, VT_);

    // in-place log_softmax per row
    log_softmax_rows<<<B_ * T_, 256, 0, stream>>>(out);
}